// Datastore_11123965296813
// MI455X (gfx1250) — compile-verified
//
#include <hip/hip_runtime.h>

typedef __attribute__((ext_vector_type(2))) float v2f;
typedef __attribute__((ext_vector_type(8))) float v8f;

#define D_EMB   768
#define KSTEPS  (D_EMB / 4)     // 192 K-steps of wmma_f32_16x16x4
#define CHUNK   2048            // training points per block
#define SUBT    (CHUNK / 16)    // 128 16-wide subtiles per chunk
#define KSEL    32              // top-k
#define MAXCH   128             // max chunks supported by phase-2 LDS

// ---------------------------------------------------------------------------
// Phase 1: fused f32-WMMA distance + per-chunk streaming top-32.
// Ranks by s = x^2 - 2*q.x (q^2 is a per-query constant, added in phase 2).
// Each wave drives TWO independent 16x16 accumulator tiles so the matrix pipe
// always has an independent WMMA to issue, and each A (query) fragment load
// is shared by two WMMAs.
// ---------------------------------------------------------------------------
__global__ __launch_bounds__(256) void knn_dist_topk(
    const float* __restrict__ Q, const float* __restrict__ X,
    int N, float* __restrict__ cand_d, int* __restrict__ cand_i, int nchunks)
{
  __shared__ float dist[16][CHUNK + 4];   // +4 pad vs 64-bank conflicts

  const int lane  = threadIdx.x & 31;
  const int wave  = threadIdx.x >> 5;
  const int chunk = blockIdx.x;
  const int qbase = blockIdx.y * 16;
  const int cbase = chunk * CHUNK;

  // init distances to +inf (covers the ragged last chunk)
  for (int i = threadIdx.x; i < 16 * (CHUNK + 4); i += blockDim.x)
    (&dist[0][0])[i] = __builtin_inff();
  __syncthreads();

  // WMMA f32 operand mapping (wave32):
  //   A 16x4: lanes 0-15 -> M=lane, K={0,1}; lanes 16-31 -> M=lane-16, K={2,3}
  //   B  4x16 mirrored; C/D: VGPR r -> rows r (lanes 0-15) and r+8 (lanes 16-31)
  const int   mrow  = lane & 15;
  const int   khalf = (lane >> 4) << 1;         // 0 or 2
  const float* qrow = Q + (size_t)(qbase + mrow) * D_EMB + khalf;

  for (int pass = 0; pass < SUBT / 16; ++pass) {    // 8 passes x 8 waves x 2 tiles
    const int st0 = pass * 16 + wave;
    const int st1 = st0 + 8;
    const int cb0 = cbase + st0 * 16;
    if (cb0 >= N) break;                            // uniform: EXEC stays full
    const int  cb1  = cbase + st1 * 16;
    const bool v1   = (cb1 < N);                    // uniform
    const int  cb1c = v1 ? cb1 : cb0;               // clamp -> in-bounds loads

    const float* xcol0 = X + (size_t)(cb0  + mrow) * D_EMB + khalf;
    const float* xcol1 = X + (size_t)(cb1c + mrow) * D_EMB + khalf;

    v8f  acc0 = {}, acc1 = {};
    float s20 = 0.f, s21 = 0.f;                     // partial x^2 per column
#pragma unroll 2
    for (int k = 0; k < KSTEPS; ++k) {
      v2f a  = *(const v2f*)(qrow  + 4 * k);        // shared A fragment
      v2f b0 = *(const v2f*)(xcol0 + 4 * k);
      v2f b1 = *(const v2f*)(xcol1 + 4 * k);
      s20 += b0.x * b0.x + b0.y * b0.y;
      s21 += b1.x * b1.x + b1.y * b1.y;
      acc0 = __builtin_amdgcn_wmma_f32_16x16x4_f32(
          false, a, false, b0, (short)0, acc0, false, false);
      acc1 = __builtin_amdgcn_wmma_f32_16x16x4_f32(
          false, a, false, b1, (short)0, acc1, false, false);
    }
    const float x20   = s20 + __shfl_xor(s20, 16, 32);
    const float x21   = s21 + __shfl_xor(s21, 16, 32);
    const int   rbase = (lane < 16) ? 0 : 8;
#pragma unroll
    for (int r = 0; r < 8; ++r)
      dist[rbase + r][st0 * 16 + mrow] = x20 - 2.0f * acc0[r];
    if (v1) {
#pragma unroll
      for (int r = 0; r < 8; ++r)
        dist[rbase + r][st1 * 16 + mrow] = x21 - 2.0f * acc1[r];
    }
  }
  __syncthreads();

  // Streaming top-32 per query: 2 queries per wave, wave-wide argmin extraction.
  for (int qq = 0; qq < 2; ++qq) {
    const int ql = wave * 2 + qq;
    const int qg = qbase + ql;
    for (int t = 0; t < KSEL; ++t) {
      float best = __builtin_inff();
      int   bcol = 0x7fffffff;
      for (int c = lane; c < CHUNK; c += 32) {
        float v = dist[ql][c];
        if (v < best) { best = v; bcol = c; }   // ascending scan => lowest idx on tie
      }
#pragma unroll
      for (int off = 16; off > 0; off >>= 1) {  // butterfly: all lanes agree
        float ov = __shfl_xor(best, off, 32);
        int   oc = __shfl_xor(bcol, off, 32);
        if (ov < best || (ov == best && oc < bcol)) { best = ov; bcol = oc; }
      }
      if (lane == 0) {
        dist[ql][bcol] = __builtin_inff();      // remove extracted element
        const size_t o = (size_t)qg * ((size_t)nchunks * KSEL) + (size_t)chunk * KSEL + t;
        cand_d[o] = best;
        cand_i[o] = cbase + bcol;
      }
    }
  }
}

// ---------------------------------------------------------------------------
// Phase 2: per-query candidate merge + linear layer + prefix softmax.
// One wave32 per query (k == 32 == wave size).
// ---------------------------------------------------------------------------
__global__ __launch_bounds__(32) void knn_finalize(
    const float* __restrict__ Q,
    const float* __restrict__ cand_d, const int* __restrict__ cand_i,
    const float* __restrict__ Y, const int* __restrict__ sys_tr,
    const int* __restrict__ qsys,
    const float* __restrict__ W, const float* __restrict__ bias,
    float* __restrict__ out, int Bq, int nchunks)
{
  __shared__ float cd[MAXCH * KSEL];
  __shared__ int   ci[MAXCH * KSEL];

  const int q    = blockIdx.x;
  const int lane = threadIdx.x;
  const int NC   = nchunks * KSEL;

  for (int i = lane; i < NC; i += 32) {
    cd[i] = cand_d[(size_t)q * NC + i];
    ci[i] = cand_i[(size_t)q * NC + i];
  }
  __syncthreads();

  // q^2 for this query (restores true squared L2)
  const float* qrow = Q + (size_t)q * D_EMB;
  float q2 = 0.f;
  for (int i = lane; i < D_EMB; i += 32) { float v = qrow[i]; q2 += v * v; }
#pragma unroll
  for (int off = 16; off > 0; off >>= 1) q2 += __shfl_xor(q2, off, 32);

  // 32 rounds of wave argmin extraction -> lane t owns neighbor t (ascending)
  float myD = 0.f; int myI = 0;
  for (int t = 0; t < KSEL; ++t) {
    float best = __builtin_inff();
    int   bgi  = 0x7fffffff;
    int   bpos = 0;
    for (int c = lane; c < NC; c += 32) {
      float v = cd[c]; int gi = ci[c];
      if (v < best || (v == best && gi < bgi)) { best = v; bgi = gi; bpos = c; }
    }
#pragma unroll
    for (int off = 16; off > 0; off >>= 1) {
      float ov = __shfl_xor(best, off, 32);
      int   og = __shfl_xor(bgi,  off, 32);
      int   op = __shfl_xor(bpos, off, 32);
      if (ov < best || (ov == best && og < bgi)) { best = ov; bgi = og; bpos = op; }
    }
    if (lane == 0) cd[bpos] = __builtin_inff();
    if (lane == t) { myD = best + q2; myI = bgi; }
  }

  // local_layer: loc_dis[j] = D*W[j,0] + b[j]; pick by sys match
  const float score = Y[myI];
  const int   lb    = (sys_tr[myI] == qsys[q]);
  const float newD  = lb ? (myD * W[1] + bias[1]) : (myD * W[0] + bias[0]);

  // prefix softmax-weighted scores (row max == every prefix max, per reference)
  const float neg = -newD;
  float m = neg;
#pragma unroll
  for (int off = 16; off > 0; off >>= 1) m = fmaxf(m, __shfl_xor(m, off, 32));
  float wv = expf(neg - m);
  float den = wv, num = wv * score;
#pragma unroll
  for (int off = 1; off < 32; off <<= 1) {      // inclusive scans
    float dn = __shfl_up(den, off, 32);
    float nm = __shfl_up(num, off, 32);
    if (lane >= off) { den += dn; num += nm; }
  }

  out[(size_t)q * KSEL + lane] = newD;                              // new_D
  out[(size_t)Bq * KSEL + (size_t)q * KSEL + lane] = num / den;     // knns_scores
}

// ---------------------------------------------------------------------------
extern "C" void kernel_launch(void* const* d_in, const int* in_sizes, int n_in,
                              void* d_out, int out_size, void* d_ws, size_t ws_size,
                              hipStream_t stream) {
  const float* Q      = (const float*)d_in[0];
  const int*   qsys   = (const int*)  d_in[1];
  const float* X      = (const float*)d_in[2];
  const float* Y      = (const float*)d_in[3];
  const int*   sys_tr = (const int*)  d_in[4];
  const float* W      = (const float*)d_in[5];
  const float* bias   = (const float*)d_in[6];

  const int Bq = in_sizes[0] / D_EMB;                 // 256
  const int N  = in_sizes[2] / D_EMB;                 // 200000
  int nchunks  = (N + CHUNK - 1) / CHUNK;             // 98
  if (nchunks > MAXCH) nchunks = MAXCH;

  float* cand_d = (float*)d_ws;
  int*   cand_i = (int*)((char*)d_ws + (size_t)Bq * nchunks * KSEL * sizeof(float));

  dim3 g1(nchunks, (Bq + 15) / 16);
  knn_dist_topk<<<g1, 256, 0, stream>>>(Q, X, N, cand_d, cand_i, nchunks);
  knn_finalize<<<Bq, 32, 0, stream>>>(Q, cand_d, cand_i, Y, sys_tr, qsys,
                                      W, bias, (float*)d_out, Bq, nchunks);
}